// FlowCPAB_23038204576183
// MI455X (gfx1250) — compile-verified
//
#include <hip/hip_runtime.h>

// CPAB flow, chain-table formulation.
// Insight: the velocity field is continuous => flow direction never reverses,
// and every trajectory that leaves its initial cell enters the next cell at a
// cell boundary. The remaining trajectory from a boundary is a universal
// function of remaining time. Only 12 boundary-entry states exist
// (5 cells x 2 sides + 2 edge "stay" states), so a 1-wave pre-kernel
// precomputes, per state, cumulative exit times T[k], cumulative logder L[k],
// and per-segment dynamics. Each point then needs: one exit-time solve
// (1 log + 1 rcp), a 7-compare table search, and one exp. ~85 VALU/point vs
// ~350 for the direct 7-step loop (kernel is VALU-bound: 5.9e9 vs 1.5e9
// lane-ops; HBM floor for 192MB is ~8us @ 23.3 TB/s).

#define NC 5
#define NSEG 8          // max chain segments (<=6 needed; padded to 8)
#define NCHAIN 12       // 5 cells x {entered-left, entered-right} + stay@1 + stay@0
#define TAB_FLOATS 512  // cellA 20 + cellB 10 + pad 2 + T 96 + seg 384

typedef float v2f __attribute__((ext_vector_type(2)));
typedef float v4f __attribute__((ext_vector_type(4)));

#define LOG2E_F 1.4426950408889634f
#define LN2_F   0.6931471805599453f
#define TINYF   1e-12f
#define BIGF    1e10f
#define EPSF    1e-7f
#define TINF    1e30f
#define ABIG    1e-8f          // |a| threshold
#define KEBIG   1.4426950e-8f  // |a*log2e| threshold

__device__ __forceinline__ float fast_log2(float v) { return __builtin_amdgcn_logf(v); }   // v_log_f32
__device__ __forceinline__ float fast_exp2(float v) { return __builtin_amdgcn_exp2f(v); }  // v_exp_f32
__device__ __forceinline__ float fast_rcp (float v) { return __builtin_amdgcn_rcpf(v);  }  // v_rcp_f32

// ---------------------------------------------------------------------------
// Pre-kernel: one wave builds the whole table into d_ws (512 floats).
// Layout (float offsets):
//   [  0.. 19] cellA[5] = {a, b, a*log2e, b/sa}
//   [ 20.. 29] cellB[5] = {ln2/sa, 1/sb}
//   [ 30.. 31] pad (zeros)
//   [ 32..127] T[12][8]   cumulative segment start times (INF-padded)
//   [128..511] seg[12][8] = {entry_x, a*log2e, b/sa, Lcum}
// ---------------------------------------------------------------------------
__global__ __launch_bounds__(32) void cpab_build_tables(
        const float* __restrict__ theta,   // 6
        const float* __restrict__ Bm,      // 10 x 6 row-major
        float* __restrict__ ws) {
    __shared__ float la[NC], lb[NC];
    int tid = threadIdx.x;

    if (tid < 2 * NC) {                    // A = B @ theta
        const float* Br = Bm + tid * (NC + 1);
        float acc = 0.0f;
#pragma unroll
        for (int j = 0; j < NC + 1; ++j) acc = fmaf(Br[j], theta[j], acc);
        if (tid & 1) lb[tid >> 1] = acc; else la[tid >> 1] = acc;
    }
    __syncthreads();

    if (tid < 2) ws[30 + tid] = 0.0f;      // pad

    if (tid >= 16 && tid < 16 + NC) {      // per-cell first-step tables
        int c = tid - 16;
        float a = la[c], b = lb[c];
        float sa = (fabsf(a) > ABIG)  ? a : 1.0f;
        float sb = (fabsf(b) > TINYF) ? b : TINYF;
        ws[c * 4 + 0] = a;
        ws[c * 4 + 1] = b;
        ws[c * 4 + 2] = a * LOG2E_F;
        ws[c * 4 + 3] = b / sa;
        ws[20 + c * 2 + 0] = LN2_F / sa;
        ws[20 + c * 2 + 1] = 1.0f / sb;
    }

    if (tid < NCHAIN) {                    // walk the 12 chains
        int s = tid;
        int mode;                          // 0=active, 1=emit stay segment, 2=done
        float xw;
        int c = 0;
        float Tcum = 0.0f, Lcum = 0.0f, wa = 0.0f, wb = 0.0f;
        if (s < 10) { c = s >> 1; xw = (float)(c + (s & 1)) * 0.2f; mode = 0; }
        else        { xw = (s == 10) ? 1.0f : 0.0f; mode = 1; }

        for (int k = 0; k < NSEG; ++k) {
            float Tw;
            if (mode == 0)      { wa = la[c]; wb = lb[c]; Tw = Tcum; }
            else if (mode == 1) { wa = 0.0f;  wb = 0.0f;  Tw = Tcum; }
            else                { Tw = TINF; }          // keep last wa/wb/xw/Lcum
            float sa = (fabsf(wa) > ABIG) ? wa : 1.0f;
            ws[32 + s * NSEG + k] = Tw;
            float* sp = ws + 128 + (s * NSEG + k) * 4;
            sp[0] = xw; sp[1] = wa * LOG2E_F; sp[2] = wb / sa; sp[3] = Lcum;

            if (mode == 1) { mode = 2; }
            else if (mode == 0) {
                bool  big   = fabsf(wa) > ABIG;
                float v     = fmaf(wa, xw, wb);
                bool  right = v >= 0.0f;
                float xb    = (float)(c + (right ? 1 : 0)) * 0.2f;
                float bsa   = wb / sa;
                float den   = xw + bsa;
                float sden  = (fabsf(den) > TINYF) ? den : TINYF;
                float ratio = (xb + bsa) / sden;
                float th_a  = fast_log2(fmaxf(ratio, TINYF)) * LN2_F / sa;
                float sb    = (fabsf(wb) > TINYF) ? wb : TINYF;
                float th_b  = (xb - xw) / sb;
                float thit  = big ? th_a : th_b;
                bool valid  = (fabsf(v) > TINYF) && (thit >= 0.0f) &&
                              ((!big) || (ratio > 0.0f));
                if (!valid) { mode = 2; }  // asymptotic: this segment persists
                else {
                    Tcum += thit;
                    Lcum  = fmaf(wa, thit, Lcum);
                    xw = xb;
                    int ncr = c + (right ? 1 : -1);
                    if (ncr < 0 || ncr > NC - 1) mode = 1;  // domain edge: stay
                    else c = ncr;
                }
            }
        }
    }
}

// ---------------------------------------------------------------------------
// Per-point evaluation against the LDS table.
// ---------------------------------------------------------------------------
__device__ __forceinline__ void cpab_point(float xr, const float* __restrict__ tab,
                                           float& zo, float& lo) {
    const v4f*   cellA = (const v4f*)tab;            // [5]
    const v2f*   cellB = (const v2f*)(tab + 20);     // [5]
    const float* Tt    = tab + 32;                   // [12][8]
    const v4f*   seg   = (const v4f*)(tab + 128);    // [12*8]

    float xc = fminf(fmaxf(xr, EPSF), 1.0f - EPSF);
    float ff = floorf(xc * (float)NC);
    ff = fminf(fmaxf(ff, 0.0f), (float)(NC - 1));
    int c = (int)ff;

    v4f cA = cellA[c];                               // ds_load_b128
    v2f cB = cellB[c];                               // ds_load_b64
    float a = cA.x, b = cA.y, ke = cA.z, bsa = cA.w;
    float ka = cB.x, rsb = cB.y;

    // First exit-time solve from interior point (reference step formulas).
    float v     = fmaf(a, xc, b);
    bool  right = v >= 0.0f;
    bool  big   = fabsf(a) > ABIG;
    float xb    = (ff + (right ? 1.0f : 0.0f)) * 0.2f;
    float den   = xc + bsa;
    float sden  = (fabsf(den) > TINYF) ? den : TINYF;
    float ratio = (xb + bsa) * fast_rcp(sden);
    float th_a  = fast_log2(fmaxf(ratio, TINYF)) * ka;   // log(r)/sa
    float th_b  = (xb - xc) * rsb;
    float thit  = big ? th_a : th_b;
    bool  valid = (fabsf(v) > TINYF) && (thit >= 0.0f) && ((!big) || (ratio > 0.0f));
    thit = valid ? thit : BIGF;
    bool  hit   = thit < 1.0f;

    // Chain lookup for the post-boundary remainder.
    float tp  = 1.0f - thit;
    float ld0 = hit ? a * thit : 0.0f;
    int   ncr = c + (right ? 1 : -1);
    bool  edge = (ncr < 0) || (ncr > NC - 1);
    int   s = edge ? (right ? 10 : 11) : (ncr * 2 + (right ? 0 : 1));

    const v4f* Trow = (const v4f*)(Tt + s * NSEG);
    v4f t0 = Trow[0];                                // ds_load_b128
    v4f t1 = Trow[1];                                // ds_load_b128
    int k = 0; float Ts = 0.0f;
    { bool g;
      g = tp >= t0.y; k += g; Ts = g ? t0.y : Ts;
      g = tp >= t0.z; k += g; Ts = g ? t0.z : Ts;
      g = tp >= t0.w; k += g; Ts = g ? t0.w : Ts;
      g = tp >= t1.x; k += g; Ts = g ? t1.x : Ts;
      g = tp >= t1.y; k += g; Ts = g ? t1.y : Ts;
      g = tp >= t1.z; k += g; Ts = g ? t1.z : Ts;
      g = tp >= t1.w; k += g; Ts = g ? t1.w : Ts;
    }
    v4f sg = seg[s * NSEG + k];                      // ds_load_b128

    // Unified final evaluation (no-hit path reuses it with the initial cell).
    float entry = hit ? sg.x : xc;
    float kef   = hit ? sg.y : ke;
    float bsaf  = hit ? sg.z : bsa;
    float Lf    = hit ? sg.w : 0.0f;
    float dtf   = hit ? (tp - Ts) : 1.0f;
    float af    = kef * LN2_F;
    bool  bigf  = fabsf(kef) > KEBIG;
    float bf    = bigf ? bsaf * af : bsaf;           // !big => bsa == b (sa=1)
    float eat   = fast_exp2(kef * dtf);
    zo = bigf ? fmaf(entry, eat, bsaf * (eat - 1.0f)) : fmaf(bf, dtf, entry);
    lo = ld0 + Lf + af * dtf;
}

__global__ __launch_bounds__(256) void cpab_flow_kernel(
        const float* __restrict__ x,
        const float* __restrict__ ws,      // 512-float table
        float* __restrict__ z_out,
        float* __restrict__ ld_out,
        int n4, int n) {
    __shared__ __align__(16) float tab[TAB_FLOATS];
    int tid = threadIdx.x;
    if (tid < TAB_FLOATS / 4)
        ((v4f*)tab)[tid] = ((const v4f*)ws)[tid];    // 2KB table -> LDS
    __syncthreads();

    int gid = blockIdx.x * blockDim.x + tid;

    if (gid < n4) {
        const v4f* xp = (const v4f*)x;
        __builtin_prefetch(xp + gid + 256, 0, 0);            // global_prefetch_b8
        v4f xv = __builtin_nontemporal_load(xp + gid);       // read-once stream

        float z0, z1, z2, z3, l0, l1, l2, l3;
        cpab_point(xv.x, tab, z0, l0);
        cpab_point(xv.y, tab, z1, l1);
        cpab_point(xv.z, tab, z2, l2);
        cpab_point(xv.w, tab, z3, l3);

        v4f zv = {z0, z1, z2, z3};
        v4f lv = {l0, l1, l2, l3};
        __builtin_nontemporal_store(zv, ((v4f*)z_out) + gid);
        __builtin_nontemporal_store(lv, ((v4f*)ld_out) + gid);
    } else if (gid == n4) {
        for (int i = n4 * 4; i < n; ++i) {                   // scalar tail
            float z, l;
            cpab_point(x[i], tab, z, l);
            z_out[i]  = z;
            ld_out[i] = l;
        }
    }
}

extern "C" void kernel_launch(void* const* d_in, const int* in_sizes, int n_in,
                              void* d_out, int out_size, void* d_ws, size_t ws_size,
                              hipStream_t stream) {
    const float* x     = (const float*)d_in[0];
    const float* theta = (const float*)d_in[1];   // (1, 6)
    const float* Bm    = (const float*)d_in[2];   // (10, 6)
    int n  = in_sizes[0];
    int n4 = n / 4;

    float* ws = (float*)d_ws;                     // 512 floats of scratch
    float* z  = (float*)d_out;
    float* ld = z + n;

    cpab_build_tables<<<1, 32, 0, stream>>>(theta, Bm, ws);

    int work  = n4 + ((n & 3) ? 1 : 0);
    int block = 256;
    int grid  = (work + block - 1) / block;
    if (grid < 1) grid = 1;
    cpab_flow_kernel<<<grid, block, 0, stream>>>(x, ws, z, ld, n4, n);
}